// AggregationLayer_48825188221362
// MI455X (gfx1250) — compile-verified
//
#include <hip/hip_runtime.h>
#include <stdint.h>

// Soft AND/OR aggregation scan over 64 columns per row.
//  - gfx1250 async global->LDS DMA (global_load_async_to_lds_b128, ASYNCcnt)
//  - row-major LDS tile padded to 68 floats/row (16B aligned b128 writes,
//    immediate-offset ds reads, only 2-way bank conflicts)
//  - base-2 transcendental math: t = log1p(exp(-10*df))/10 computed as
//    v_log_f32(1 + v_exp_f32(df * -10*log2(e))) * (ln2/10), with the
//    (2q-1)*ln2/10 factor pre-folded per column.

#define NCOL          64
#define TILE_ROWS     256
#define ROW_STRIDE_F  68u                      // floats: 272 B, 16B aligned
#define ROW_STRIDE_B  (ROW_STRIDE_F * 4u)

__global__ __launch_bounds__(TILE_ROWS)
void agg_scan_kernel(const float* __restrict__ x,
                     const float* __restrict__ w,
                     float* __restrict__ out,
                     int B)
{
    __shared__ float  tile[TILE_ROWS * ROW_STRIDE_F];   // ~68 KB
    __shared__ float2 qsh[NCOL - 1];

    const unsigned tid = threadIdx.x;

    // Low 32 bits of generic LDS pointer == byte offset in the workgroup's LDS.
    const unsigned lds_base = (unsigned)(uintptr_t)(&tile[0]);

    // ---- Stage tile: 16x async b128 per thread, constant address increments.
    // Global linear byte o = tid*16 + it*4096 ; row = o>>8 ; within-row = o&255.
    // LDS dst = row*272 + (o&255): per-thread base + it*4352.
    const unsigned laddr0 =
        lds_base + (tid >> 4) * ROW_STRIDE_B + (tid & 15u) * 16u;
    const unsigned long long xaddr = (unsigned long long)(uintptr_t)x;
    const unsigned long long ga0 =
        xaddr + (unsigned long long)blockIdx.x * (TILE_ROWS * NCOL * 4ull)
              + (unsigned long long)tid * 16ull;

    const size_t baseRow = (size_t)blockIdx.x * TILE_ROWS;
    if (baseRow + TILE_ROWS <= (size_t)B) {
        // full tile: no clamping, pure constant-stride issue
        #pragma unroll
        for (unsigned it = 0; it < 16u; ++it) {
            unsigned           la = laddr0 + it * (16u * ROW_STRIDE_B);
            unsigned long long ga = ga0 + (unsigned long long)it * 4096ull;
            asm volatile("global_load_async_to_lds_b128 %0, %1, off"
                         :: "v"(la), "v"(ga) : "memory");
        }
    } else {
        // tail block: clamp global address (16B aligned), LDS dsts stay unique
        const unsigned long long gmax =
            xaddr + (unsigned long long)B * (NCOL * 4ull) - 16ull;
        #pragma unroll
        for (unsigned it = 0; it < 16u; ++it) {
            unsigned           la = laddr0 + it * (16u * ROW_STRIDE_B);
            unsigned long long ga = ga0 + (unsigned long long)it * 4096ull;
            if (ga > gmax) ga = gmax;
            asm volatile("global_load_async_to_lds_b128 %0, %1, off"
                         :: "v"(la), "v"(ga) : "memory");
        }
    }

    // ---- Overlap with DMA: per-column coefficients from softmax pair.
    // q = p_or = sigmoid(w1 - w0);  score' = mn + q*df + (2q-1)*t
    if (tid < NCOL - 1) {
        float d = w[2u * tid] - w[2u * tid + 1u];
        float q = 1.0f / (1.0f + __expf(d));
        qsh[tid] = make_float2(q, (2.0f * q - 1.0f) * 0.0693147180559945f);
    }

    asm volatile("s_wait_asynccnt 0x0" ::: "memory");
    __syncthreads();

    // ---- Sequential soft AND/OR scan (registers + immediate-offset LDS reads)
    const float* rowp = &tile[tid * ROW_STRIDE_F];
    float s = rowp[0];

    #pragma unroll
    for (unsigned cc = 1; cc < NCOL; ++cc) {
        float xi = rowp[cc];
        float mn = fminf(s, xi);
        float mx = fmaxf(s, xi);
        float df = mx - mn;
        // u = exp(-10*df) = 2^(df * -10*log2(e));  t2 = log2(1+u)
        float u  = __builtin_amdgcn_exp2f(df * -14.426950408889634f);
        float t2 = __builtin_amdgcn_logf(1.0f + u);
        float2 qc = qsh[cc - 1];                 // LDS broadcast (b64)
        // s = mn + q*df + (2q-1)*(ln2/10)*log2(1+u)
        s = fmaf(qc.y, t2, fmaf(qc.x, df, mn));
    }

    size_t row = baseRow + tid;
    if (row < (size_t)B) out[row] = s;
}

extern "C" void kernel_launch(void* const* d_in, const int* in_sizes, int n_in,
                              void* d_out, int out_size, void* d_ws, size_t ws_size,
                              hipStream_t stream)
{
    const float* x = (const float*)d_in[0];     // [B, 64] fp32
    const float* w = (const float*)d_in[1];     // [63, 2] fp32
    float* out = (float*)d_out;                 // [B, 1]  fp32

    int B = in_sizes[0] / NCOL;
    if (B <= 0) return;

    int grid = (B + TILE_ROWS - 1) / TILE_ROWS;
    hipLaunchKernelGGL(agg_scan_kernel, dim3(grid), dim3(TILE_ROWS), 0, stream,
                       x, w, out, B);
}